// LinearAttention_18915035971881
// MI455X (gfx1250) — compile-verified
//
#include <hip/hip_runtime.h>

// ---------------------------------------------------------------------------
// Linear attention (elu+1 feature map) for MI455X / gfx1250, wave32 + WMMA.
//
//   KV[n,h,d,m] = sum_s fm(K[n,s,h,d]) * V[n,s,h,m]
//   Ksum[n,h,d] = sum_s fm(K[n,s,h,d])
//   Z[n,l,h]    = 1/(fm(Q[n,l,h,:]) . Ksum[n,h,:]) + 1e-10
//   out[n,l,h,m]= Z * sum_d fm(Q[n,l,h,d]) * KV[n,h,d,m]
//
// Both GEMMs use V_WMMA_F32_16X16X4_F32 (fp32 A/B/C -> no precision loss).
// Memory-bound problem (512 MB total traffic vs ~8.6 GFLOP), so the design
// streams every tensor exactly once from HBM.
// ---------------------------------------------------------------------------

typedef __attribute__((ext_vector_type(2))) float v2f;
typedef __attribute__((ext_vector_type(8))) float v8f;

#define N_ 4
#define H_ 16
#define L_ 8192
#define D_ 64
#define SSTRIDE (H_ * D_)          // 1024 floats between consecutive seq rows
#define CHUNK 256                  // L-rows per wave in pass 1
#define NCHUNKS (L_ / CHUNK)       // 32
#define TILES_PER_WAVE 8           // 8 * 16 = 128 L-rows per wave in pass 2
#define QPAD 68                    // LDS row pitch (floats), conflict-free

__device__ __forceinline__ float feature_map(float x) {
  // elu(x) + 1 == (x > 0 ? x + 1 : exp(x))
  return x > 0.0f ? x + 1.0f : __expf(x);
}

// D = A(16x4 f32) * B(4x16 f32) + C(16x16 f32)
// A layout (ISA 7.12.2): lanes 0-15 hold row M=lane, K={0,1} in .x/.y;
//                        lanes 16-31 hold row M=lane-16, K={2,3}.
// B layout assumed analogous: lanes 0-15 -> K={0,1}, lanes 16-31 -> K={2,3},
//                             column N = lane&15, rows in .x/.y.
// C/D layout: VGPR r, lanes 0-15 -> (M=r,    N=lane),
//                     lanes 16-31 -> (M=r+8,  N=lane-16).
__device__ __forceinline__ v8f wmma_f32_16x16x4(v2f a, v2f b, v8f c) {
  return __builtin_amdgcn_wmma_f32_16x16x4_f32(false, a, false, b,
                                               (short)0, c, false, false);
}

// ---------------------------------------------------------------------------
// Zero-fill workspace (KV + Ksum accumulators) -- required before atomics.
// ---------------------------------------------------------------------------
__global__ __launch_bounds__(256) void la_zero(float* p, int n) {
  int i = blockIdx.x * 256 + threadIdx.x;
  if (i < n) p[i] = 0.0f;
}

// ---------------------------------------------------------------------------
// Pass 1: KV[d][m] += fm(K)^T V over an L-chunk; Ksum via VALU side-channel.
// One wave per (n, h, chunk). 16 WMMA accumulators cover the full 64x64 tile.
// ---------------------------------------------------------------------------
__global__ __launch_bounds__(256) void la_pass1(const float* __restrict__ K,
                                                const float* __restrict__ V,
                                                float* __restrict__ KVbuf,
                                                float* __restrict__ Ksum) {
  const int lane  = threadIdx.x & 31;
  const int wv    = threadIdx.x >> 5;
  const int gwave = blockIdx.x * 8 + wv;          // grid sized exactly

  const int chunk = gwave % NCHUNKS;
  const int nh    = gwave / NCHUNKS;              // 0..63
  const int h     = nh % H_;
  const int n     = nh / H_;

  const int r15  = lane & 15;                     // row within 16-tile
  const int sSel = (lane >> 4) << 1;              // 0 or 2 (K split)

  const size_t base = ((size_t)n * L_ * H_ + (size_t)h) * D_
                    + (size_t)chunk * CHUNK * SSTRIDE;
  const float* Kp = K + base;
  const float* Vp = V + base;

  const v8f zero8 = {0.f, 0.f, 0.f, 0.f, 0.f, 0.f, 0.f, 0.f};
  v8f acc[4][4];                                  // [d-tile][m-tile]
#pragma unroll
  for (int dt = 0; dt < 4; ++dt)
#pragma unroll
    for (int mt = 0; mt < 4; ++mt) acc[dt][mt] = zero8;

  float ksum[4] = {0.f, 0.f, 0.f, 0.f};

  for (int s0 = 0; s0 < CHUNK; s0 += 4) {
    const float* kr0 = Kp + (size_t)(s0 + sSel) * SSTRIDE;
    const float* kr1 = kr0 + SSTRIDE;
    const float* vr0 = Vp + (size_t)(s0 + sSel) * SSTRIDE;
    const float* vr1 = vr0 + SSTRIDE;

    v2f a[4], b[4];
#pragma unroll
    for (int t = 0; t < 4; ++t) {
      const int d = t * 16 + r15;
      float x0 = feature_map(kr0[d]);
      float x1 = feature_map(kr1[d]);
      a[t].x = x0; a[t].y = x1;
      ksum[t] += x0 + x1;
      b[t].x = vr0[d];                            // m column == t*16+r15
      b[t].y = vr1[d];
    }
#pragma unroll
    for (int dt = 0; dt < 4; ++dt)
#pragma unroll
      for (int mt = 0; mt < 4; ++mt)
        acc[dt][mt] = wmma_f32_16x16x4(a[dt], b[mt], acc[dt][mt]);
  }

  // Merge partial KV tile into L2-resident accumulator buffer.
  float* kvDst = KVbuf + (size_t)nh * D_ * D_;
  const int rowAdd = (lane >> 4) << 3;            // +8 for upper half-wave
#pragma unroll
  for (int dt = 0; dt < 4; ++dt)
#pragma unroll
    for (int mt = 0; mt < 4; ++mt)
#pragma unroll
      for (int r = 0; r < 8; ++r) {
        const int d = dt * 16 + r + rowAdd;
        const int m = mt * 16 + r15;
        atomicAdd(&kvDst[d * D_ + m], acc[dt][mt][r]);
      }

  float* ksDst = Ksum + (size_t)nh * D_;
#pragma unroll
  for (int t = 0; t < 4; ++t)
    atomicAdd(&ksDst[t * 16 + r15], ksum[t]);
}

// ---------------------------------------------------------------------------
// Pass 2: out = Z * fm(Q) @ KV. One wave per 128 L-rows of one (n,h).
// KV B-fragments are preloaded once into 128 VGPRs and reused across 8 tiles.
// Q rows are staged coalesced into LDS (feature map applied), then read as
// strided A-fragments (pad 68 -> bank-conflict free).
// ---------------------------------------------------------------------------
__global__ __launch_bounds__(256) void la_pass2(const float* __restrict__ Q,
                                                const float* __restrict__ KVbuf,
                                                const float* __restrict__ Ksum,
                                                float* __restrict__ out) {
  __shared__ float qs_all[8 * 16 * QPAD];
  __shared__ float ks[D_];

  const int lane  = threadIdx.x & 31;
  const int wv    = threadIdx.x >> 5;
  const int gwave = blockIdx.x * 8 + wv;

  const int wavesPerNH = L_ / (16 * TILES_PER_WAVE); // 64, divisible by 8:
  const int nh    = gwave / wavesPerNH;              // -> whole block same nh
  const int wslot = gwave % wavesPerNH;
  const int h     = nh % H_;
  const int n     = nh / H_;
  const int lbase0 = wslot * (16 * TILES_PER_WAVE);

  const int r15  = lane & 15;
  const int sSel = (lane >> 4) << 1;
  const int rowAdd = (lane >> 4) << 3;

  // Stage Ksum for this (n,h) once per block.
  if (threadIdx.x < D_) ks[threadIdx.x] = Ksum[(size_t)nh * D_ + threadIdx.x];
  __syncthreads();

  // Preload all B fragments of the 64x64 KV tile (L2-resident, 16 KB).
  const float* kv = KVbuf + (size_t)nh * D_ * D_;
  v2f Bf[16][4];
#pragma unroll
  for (int kd = 0; kd < 16; ++kd)
#pragma unroll
    for (int mt = 0; mt < 4; ++mt) {
      const int d0 = kd * 4 + sSel;
      Bf[kd][mt].x = kv[d0 * D_ + mt * 16 + r15];
      Bf[kd][mt].y = kv[(d0 + 1) * D_ + mt * 16 + r15];
    }

  float* qs = qs_all + wv * 16 * QPAD;
  const v8f zero8 = {0.f, 0.f, 0.f, 0.f, 0.f, 0.f, 0.f, 0.f};

  for (int t = 0; t < TILES_PER_WAVE; ++t) {
    const int lbase = lbase0 + t * 16;
    const size_t rowOff = (((size_t)n * L_ + lbase) * H_ + h) * D_;

    // Coalesced stage of 16 Q rows with feature map applied.
    for (int i = lane; i < 16 * D_; i += 32) {
      const int r = i >> 6, d = i & 63;
      qs[r * QPAD + d] = feature_map(Q[rowOff + (size_t)r * SSTRIDE + d]);
    }
    __syncthreads();

    // Z normalizer: lane computes row (lane&15); lanes 16-31 duplicate.
    float dot = 0.f;
#pragma unroll 8
    for (int d = 0; d < D_; ++d) dot += qs[r15 * QPAD + d] * ks[d];
    const float z = 1.0f / dot + 1e-10f;

    // GEMM: 16 k-steps over D, 4 m-tiles.
    v8f acc[4] = {zero8, zero8, zero8, zero8};
#pragma unroll
    for (int kd = 0; kd < 16; ++kd) {
      v2f a;
      a.x = qs[r15 * QPAD + kd * 4 + sSel];
      a.y = qs[r15 * QPAD + kd * 4 + sSel + 1];
#pragma unroll
      for (int mt = 0; mt < 4; ++mt)
        acc[mt] = wmma_f32_16x16x4(a, Bf[kd][mt], acc[mt]);
    }

    // Scale by per-row Z (shuffle broadcast) and store.
    float* obase = out + rowOff;
#pragma unroll
    for (int r = 0; r < 8; ++r) {
      const int row = r + rowAdd;
      const float zr = __shfl(z, row, 32);
#pragma unroll
      for (int mt = 0; mt < 4; ++mt)
        obase[(size_t)row * SSTRIDE + mt * 16 + r15] = acc[mt][r] * zr;
    }
    __syncthreads();   // protect qs before next tile overwrites it
  }
}

// ---------------------------------------------------------------------------
extern "C" void kernel_launch(void* const* d_in, const int* in_sizes, int n_in,
                              void* d_out, int out_size, void* d_ws,
                              size_t ws_size, hipStream_t stream) {
  (void)in_sizes; (void)n_in; (void)out_size; (void)ws_size;

  const float* q = (const float*)d_in[0];
  const float* k = (const float*)d_in[1];
  const float* v = (const float*)d_in[2];
  float* outp = (float*)d_out;

  float* kvbuf = (float*)d_ws;                       // 64 * 64 * 64 floats
  float* ksum  = kvbuf + (size_t)N_ * H_ * D_ * D_;  // + 64 * 64 floats

  const int initN = N_ * H_ * D_ * D_ + N_ * H_ * D_;   // 266240 floats
  la_zero<<<(initN + 255) / 256, 256, 0, stream>>>(kvbuf, initN);

  // Pass 1: N*H*NCHUNKS = 2048 waves, 8 waves / 256-thread block.
  la_pass1<<<(N_ * H_ * NCHUNKS) / 8, 256, 0, stream>>>(k, v, kvbuf, ksum);

  // Pass 2: N*H*(L/128) = 4096 waves, 8 waves / block (all same nh per block).
  la_pass2<<<(N_ * H_ * (L_ / (16 * TILES_PER_WAVE))) / 8, 256, 0, stream>>>(
      q, kvbuf, ksum, outp);
}